// OrientationBFN_56985626084110
// MI455X (gfx1250) — compile-verified
//
#include <hip/hip_runtime.h>
#include <math.h>

// ---------------------------------------------------------------------------
// CDNA5 async LDS<->global DMA. Load builtin confirmed present with signature
// (v4i AS1*, v4i AS3*, imm int offset, imm int cpol); store assumed symmetric.
// ---------------------------------------------------------------------------
#if defined(__HIP_DEVICE_COMPILE__) && __has_builtin(__builtin_amdgcn_global_load_async_to_lds_b128)
#define USE_ASYNC 1
#else
#define USE_ASYNC 0
#endif

#if USE_ASYNC
#if __has_builtin(__builtin_amdgcn_s_wait_asynccnt)
#define WAIT_ASYNC0() __builtin_amdgcn_s_wait_asynccnt(0)
#else
#define WAIT_ASYNC0() asm volatile("s_wait_asynccnt 0" ::: "memory")
#endif
#if __has_builtin(__builtin_amdgcn_global_store_async_from_lds_b128)
#define USE_ASYNC_STORE 1
#else
#define USE_ASYNC_STORE 0
#endif
#define AS1 __attribute__((address_space(1)))
#define AS3 __attribute__((address_space(3)))
typedef int v4i __attribute__((ext_vector_type(4)));
#else
#define USE_ASYNC_STORE 0
#endif

// ---------------------------------------------------------------------------
// One approximate-Givens Jacobi rotation on symmetric 3x3 S (pivot p,q; r is
// the remaining index).  Accumulates the rotation into V columns vp, vq.
// Branch-free (one v_rsq, selects only).
// ---------------------------------------------------------------------------
__device__ __forceinline__ void jrot(float &spp, float &sqq, float &spq,
                                     float &spr, float &sqr,
                                     float *vp, float *vq)
{
    const float GAMMA = 5.828427124746190f;   // 3 + 2*sqrt(2)
    const float CSTAR = 0.923879532511287f;   // cos(pi/8)
    const float SSTAR = 0.382683432365090f;   // sin(pi/8)
    float ch = 2.0f * (spp - sqq);
    float sh = spq;
    bool  big = (GAMMA * sh * sh) < (ch * ch);
    float w   = __frsqrt_rn(ch * ch + sh * sh);
    float chn = big ? (w * ch) : CSTAR;
    float shn = big ? (w * sh) : SSTAR;
    float c = chn * chn - shn * shn;          // cos(theta)
    float s = 2.0f * chn * shn;               // sin(theta)
    float cc = c * c, ss = s * s, cs = c * s;
    float npp = cc * spp + 2.0f * cs * spq + ss * sqq;
    float nqq = ss * spp - 2.0f * cs * spq + cc * sqq;
    float npq = (cc - ss) * spq + cs * (sqq - spp);
    spp = npp; sqq = nqq; spq = npq;
    float pr = spr, qr = sqr;
    spr =  c * pr + s * qr;
    sqr = -s * pr + c * qr;
#pragma unroll
    for (int i = 0; i < 3; ++i) {
        float p = vp[i], q = vq[i];
        vp[i] =  c * p + s * q;
        vq[i] = -s * p + c * q;
    }
}

// ---------------------------------------------------------------------------
// One block = 256 matrices = 9216 B per input array (27648 B LDS tile).
// In:  async DMA global->LDS (ASYNCcnt), stride-9 dword LDS reads
//      (9 mod 64 banks -> conflict-free).
// Out: results staged back into the first LDS tile (each thread overwrites
//      only the region it already consumed), then async DMA LDS->global in
//      512B-per-instruction fully coalesced b128 chunks (full-line HBM writes).
// ---------------------------------------------------------------------------
__global__ __launch_bounds__(256) void
OrientationBFN_proj_kernel(const float* __restrict__ Rc,
                           const float* __restrict__ tbuf,
                           const float* __restrict__ no,
                           const float* __restrict__ ep,
                           float* __restrict__ out,
                           int blocks_per_n)
{
    const int tid = threadIdx.x;
    const long long base = (long long)blockIdx.x * (256LL * 9LL);
    const int n = blockIdx.x / blocks_per_n;        // uniform -> scalar load of t
    const float alpha  = tbuf[n];                   // BETA == 1.0
    const float salpha = sqrtf(alpha);
    const float JIT = 1e-6f;

    float a[9];

#if USE_ASYNC
    __shared__ float lRc[2304];
    __shared__ float lNo[2304];
    __shared__ float lEp[2304];
    // 576 float4 per array; async DMA to LDS (ASYNCcnt), bypassing VGPR returns
    for (int j = tid; j < 576; j += 256) {
        const long long go = base + (long long)j * 4;
        __builtin_amdgcn_global_load_async_to_lds_b128(
            (AS1 v4i*)(Rc + go), (AS3 v4i*)(lRc + j * 4), 0, 0);
        __builtin_amdgcn_global_load_async_to_lds_b128(
            (AS1 v4i*)(no + go), (AS3 v4i*)(lNo + j * 4), 0, 0);
        __builtin_amdgcn_global_load_async_to_lds_b128(
            (AS1 v4i*)(ep + go), (AS3 v4i*)(lEp + j * 4), 0, 0);
    }
    WAIT_ASYNC0();        // this wave's async ops done
    __syncthreads();      // all waves' tiles visible
#pragma unroll
    for (int k = 0; k < 9; ++k) {
        a[k] = fmaf(alpha, lRc[tid * 9 + k],
               fmaf(salpha, lNo[tid * 9 + k], JIT * lEp[tid * 9 + k]));
    }
#else
    const float* pR = Rc + base + tid * 9;
    const float* pN = no + base + tid * 9;
    const float* pE = ep + base + tid * 9;
#pragma unroll
    for (int k = 0; k < 9; ++k) {
        // streaming footprint (302MB) > L2 (192MB): non-temporal
        float r0 = __builtin_nontemporal_load(pR + k);
        float n0 = __builtin_nontemporal_load(pN + k);
        float e0 = __builtin_nontemporal_load(pE + k);
        a[k] = fmaf(alpha, r0, fmaf(salpha, n0, JIT * e0));
    }
#endif

    // ---- S = A^T A (columns of A: c0=(a0,a3,a6), c1=(a1,a4,a7), c2=(a2,a5,a8))
    float s00 = a[0]*a[0] + a[3]*a[3] + a[6]*a[6];
    float s01 = a[0]*a[1] + a[3]*a[4] + a[6]*a[7];
    float s02 = a[0]*a[2] + a[3]*a[5] + a[6]*a[8];
    float s11 = a[1]*a[1] + a[4]*a[4] + a[7]*a[7];
    float s12 = a[1]*a[2] + a[4]*a[5] + a[7]*a[8];
    float s22 = a[2]*a[2] + a[5]*a[5] + a[8]*a[8];

    // V column-major: V[c*3 + r], init identity
    float V[9] = {1.f,0.f,0.f, 0.f,1.f,0.f, 0.f,0.f,1.f};

#pragma unroll
    for (int sweep = 0; sweep < 4; ++sweep) {
        jrot(s00, s11, s01, s02, s12, &V[0], &V[3]);   // pivot (0,1), r=2
        jrot(s00, s22, s02, s01, s12, &V[0], &V[6]);   // pivot (0,2), r=1
        jrot(s11, s22, s12, s01, s02, &V[3], &V[6]);   // pivot (1,2), r=0
    }

    // ---- pick eigenvectors of the two LARGEST eigenvalues (drop argmin col)
    const float l0 = s00, l1 = s11, l2 = s22;
    const int imin = (l0 <= l1) ? ((l0 <= l2) ? 0 : 2) : ((l1 <= l2) ? 1 : 2);
    float v1[3], v2[3];
#pragma unroll
    for (int i = 0; i < 3; ++i) {
        v1[i] = (imin == 0) ? V[3 + i] : V[i];
        v2[i] = (imin == 2) ? V[3 + i] : V[6 + i];
    }
    float la = (imin == 0) ? l1 : l0;
    float lb = (imin == 2) ? l1 : l2;
    const bool sw = lb > la;                 // make v1 <-> largest sigma
#pragma unroll
    for (int i = 0; i < 3; ++i) {
        float t1 = v1[i], t2 = v2[i];
        v1[i] = sw ? t2 : t1;
        v2[i] = sw ? t1 : t2;
    }

    // ---- u1 = norm(A v1); u2 = GS-orth(A v2); u3 = u1 x u2; w3 = v1 x v2
    float b1x = a[0]*v1[0] + a[1]*v1[1] + a[2]*v1[2];
    float b1y = a[3]*v1[0] + a[4]*v1[1] + a[5]*v1[2];
    float b1z = a[6]*v1[0] + a[7]*v1[1] + a[8]*v1[2];
    float inv1 = __frsqrt_rn(b1x*b1x + b1y*b1y + b1z*b1z + 1e-30f);
    float u1x = b1x*inv1, u1y = b1y*inv1, u1z = b1z*inv1;

    float b2x = a[0]*v2[0] + a[1]*v2[1] + a[2]*v2[2];
    float b2y = a[3]*v2[0] + a[4]*v2[1] + a[5]*v2[2];
    float b2z = a[6]*v2[0] + a[7]*v2[1] + a[8]*v2[2];
    float d = b2x*u1x + b2y*u1y + b2z*u1z;
    b2x -= d*u1x; b2y -= d*u1y; b2z -= d*u1z;
    float inv2 = __frsqrt_rn(b2x*b2x + b2y*b2y + b2z*b2z + 1e-30f);
    float u2x = b2x*inv2, u2y = b2y*inv2, u2z = b2z*inv2;

    float u3x = u1y*u2z - u1z*u2y;
    float u3y = u1z*u2x - u1x*u2z;
    float u3z = u1x*u2y - u1y*u2x;

    float w3x = v1[1]*v2[2] - v1[2]*v2[1];
    float w3y = v1[2]*v2[0] - v1[0]*v2[2];
    float w3z = v1[0]*v2[1] - v1[1]*v2[0];

    // ---- R = u1 v1^T + u2 v2^T + u3 w3^T  (det(R)=+1, sign-corrected proj)
    float r[9];
    r[0] = u1x*v1[0] + u2x*v2[0] + u3x*w3x;
    r[1] = u1x*v1[1] + u2x*v2[1] + u3x*w3y;
    r[2] = u1x*v1[2] + u2x*v2[2] + u3x*w3z;
    r[3] = u1y*v1[0] + u2y*v2[0] + u3y*w3x;
    r[4] = u1y*v1[1] + u2y*v2[1] + u3y*w3y;
    r[5] = u1y*v1[2] + u2y*v2[2] + u3y*w3z;
    r[6] = u1z*v1[0] + u2z*v2[0] + u3z*w3x;
    r[7] = u1z*v1[1] + u2z*v2[1] + u3z*w3y;
    r[8] = u1z*v1[2] + u2z*v2[2] + u3z*w3z;

#if USE_ASYNC && USE_ASYNC_STORE
    // Stage results into the already-consumed lRc tile (each thread writes only
    // the region it itself read earlier -> no cross-thread hazard before the
    // barrier), then bulk-DMA LDS->global: 512B contiguous per instruction.
#pragma unroll
    for (int k = 0; k < 9; ++k)
        lRc[tid * 9 + k] = r[k];
    __syncthreads();      // ds stores of all waves visible before async DMA
    for (int j = tid; j < 576; j += 256) {
        const long long go = base + (long long)j * 4;
        __builtin_amdgcn_global_store_async_from_lds_b128(
            (AS1 v4i*)(out + go), (AS3 v4i*)(lRc + j * 4), 0, 0);
    }
    // S_ENDPGM performs an implicit wait-idle (all ASYNCcnt drained).
#else
    float* po = out + base + tid * 9;
#pragma unroll
    for (int k = 0; k < 9; ++k)
        __builtin_nontemporal_store(r[k], po + k);
#endif
}

extern "C" void kernel_launch(void* const* d_in, const int* in_sizes, int n_in,
                              void* d_out, int out_size, void* d_ws, size_t ws_size,
                              hipStream_t stream) {
    const float* Rc = (const float*)d_in[0];   // (N,L,3,3) f32
    const float* t  = (const float*)d_in[1];   // (N,)      f32
    const float* no = (const float*)d_in[2];   // (N,L,3,3) f32
    const float* ep = (const float*)d_in[3];   // (N,L,3,3) f32
    float* out = (float*)d_out;

    const int nmat   = in_sizes[0] / 9;        // N*L
    const int N      = in_sizes[1];            // 256
    const int blocks = nmat / 256;             // 8192 (exact for this problem)
    const int bpn    = blocks / N;             // blocks per batch index n

    hipLaunchKernelGGL(OrientationBFN_proj_kernel, dim3(blocks), dim3(256), 0,
                       stream, Rc, t, no, ep, out, bpn);
}